// Linear_18004502905034
// MI455X (gfx1250) — compile-verified
//
#include <hip/hip_runtime.h>
#include <cstdint>

typedef float v2f __attribute__((ext_vector_type(2)));
typedef float v8f __attribute__((ext_vector_type(8)));
typedef unsigned int v4u __attribute__((ext_vector_type(4)));
typedef int v8i __attribute__((ext_vector_type(8)));
typedef int v4i __attribute__((ext_vector_type(4)));

constexpr int M_DIM   = 32;
constexpr int I_DIM   = 7168;
constexpr int O_DIM   = 16384;
constexpr int BLK     = 128;                    // scale block = O tile = K chunk
constexpr int KSPLIT  = 4;
constexpr int NB_PER  = (I_DIM / BLK) / KSPLIT; // 14 K-chunks per split
constexpr int WST     = BLK + 4;                // padded LDS stride (132 floats)
constexpr int THREADS = 256;                    // 8 waves (wave32)

constexpr int WBUF = 128 * WST;                 // floats per W LDS buffer
constexpr int XBUF = 32 * WST;                  // floats per X LDS buffer
constexpr int LDS_FLOATS = 2 * WBUF + 2 * XBUF; // double-buffered

// ---------------------------------------------------------------------------
// TDM: issue one 2-D tensor_load_to_lds. Descriptor per ISA cdna5 ch.8:
//   group0: [1:0]=count(1) | lds_addr | global_addr[56:0] | type=2
//   group1: data_size=4B, pad_enable (4 DWORD pad every 128 DWORDs -> LDS
//           row stride 132 floats), tensor dims/strides, tile dims
// Groups 2/3 are zero (2-D tile; dims marked unused by tile_dim=0).
// ---------------------------------------------------------------------------
__device__ __forceinline__ void tdm_load_2d(const void* gaddr, unsigned lds_byte,
                                            unsigned td0, unsigned td1,
                                            unsigned tile0, unsigned tile1,
                                            unsigned long long stride0) {
  const unsigned long long ga = (unsigned long long)(uintptr_t)gaddr;
  v4u g0;
  g0.x = 1u;                                        // count=1, user descriptor
  g0.y = lds_byte;                                  // LDS byte address
  g0.z = (unsigned)ga;                              // global_addr[31:0]
  g0.w = (unsigned)((ga >> 32) & 0x1FFFFFFu)        // global_addr[56:32]
       | (2u << 30);                                // type = 2 ("image")
  v8i g1;
  g1[0] = (int)((2u << 16)                          // data_size = 4 bytes
              | (1u << 20)                          // pad_enable
              | (6u << 22)                          // pad_interval: 128 DWORDs
              | (3u << 25));                        // pad_amount:   4 DWORDs
  g1[1] = (int)((td0 & 0xFFFFu) << 16);             // tensor_dim0[15:0]
  g1[2] = (int)((td0 >> 16) | ((td1 & 0xFFFFu) << 16));
  g1[3] = (int)((td1 >> 16) | (tile0 << 16));       // tile_dim0
  g1[4] = (int)(tile1 & 0xFFFFu);                   // tile_dim1 (tile_dim2=0)
  g1[5] = (int)(unsigned)stride0;                   // tensor_dim0_stride lo
  g1[6] = (int)(unsigned)((stride0 >> 32) & 0xFFFFu);
  g1[7] = 0;
  v4i z4 = {0, 0, 0, 0};
#if defined(__clang_major__) && __clang_major__ >= 23
  v8i z8 = {0, 0, 0, 0, 0, 0, 0, 0};
  __builtin_amdgcn_tensor_load_to_lds(g0, g1, z4, z4, z8, 0);
#else
  __builtin_amdgcn_tensor_load_to_lds(g0, g1, z4, z4, 0);
#endif
}

__global__ __launch_bounds__(THREADS)
void gemm_dq_wmma(const float* __restrict__ x,
                  const float* __restrict__ w,
                  const float* __restrict__ sinv,
                  float* __restrict__ y) {
  extern __shared__ float lds[];
  float* Wl = lds;              // [2][128][WST]
  float* Xl = lds + 2 * WBUF;   // [2][32][WST]

  const int tid = threadIdx.x;
  const int ob  = blockIdx.x;                 // O block index (0..127)
  const long o0 = (long)ob * BLK;
  const int ib0 = blockIdx.y * NB_PER;        // K-split base chunk

  // ---- WMMA lane mapping (ISA 7.12.2, 32-bit operands, wave32) ----
  const int wave   = tid >> 5;        // 0..7 : owns O columns [16w,16w+16)
  const int lane   = tid & 31;
  const int lane16 = lane & 15;
  const int khi    = (lane >> 4) * 2; // lanes 16..31 hold K+2/K+3

  // Per-wave TDM slices: wave w DMAs W rows [16w,16w+16) and X rows [4w,4w+4)
  // of the current K chunk. Descriptor fields are wave-uniform -> SGPRs.
  auto issue_chunk = [&](int ib, int p) {
    const long kbase = (long)(ib0 + ib) * BLK;
    const float* wg = w + (o0 + wave * 16) * (long)I_DIM + kbase;
    tdm_load_2d(wg, (unsigned)((p * WBUF + wave * 16 * WST) * 4),
                (unsigned)I_DIM, (unsigned)O_DIM, BLK, 16,
                (unsigned long long)I_DIM);
    const float* xg = x + (long)(wave * 4) * I_DIM + kbase;
    tdm_load_2d(xg, (unsigned)((2 * WBUF + p * XBUF + wave * 4 * WST) * 4),
                (unsigned)I_DIM, (unsigned)M_DIM, BLK, 4,
                (unsigned long long)I_DIM);
  };

  v8f acc0 = {}; // M = 0..15 tile
  v8f acc1 = {}; // M = 16..31 tile

  issue_chunk(0, 0);
  __builtin_amdgcn_s_wait_tensorcnt(0);
  __syncthreads();

  int p = 0;
  for (int ib = 0; ib < NB_PER; ++ib) {
    if (ib + 1 < NB_PER) issue_chunk(ib + 1, p ^ 1);  // DMA next chunk

    const float s = sinv[(long)ob * (I_DIM / BLK) + (ib0 + ib)];

    const float* Xb0 = Xl + p * XBUF + lane16 * WST + khi;
    const float* Xb1 = Xb0 + 16 * WST;
    const float* Bb  = Wl + p * WBUF + (wave * 16 + lane16) * WST + khi;

    v8f p0 = {}; v8f p1 = {};
#pragma unroll
    for (int kk = 0; kk < BLK; kk += 4) {
      v2f a0 = *(const v2f*)(Xb0 + kk);   // A frag, rows 0..15
      v2f a1 = *(const v2f*)(Xb1 + kk);   // A frag, rows 16..31
      v2f b  = *(const v2f*)(Bb + kk);    // B frag, 4x16 (N = lane%16)
      p0 = __builtin_amdgcn_wmma_f32_16x16x4_f32(false, a0, false, b,
                                                 (short)0, p0, false, false);
      p1 = __builtin_amdgcn_wmma_f32_16x16x4_f32(false, a1, false, b,
                                                 (short)0, p1, false, false);
    }
    // fold per-(o-block,i-block) dequant scale
    acc0 += s * p0;
    acc1 += s * p1;

    __builtin_amdgcn_s_wait_tensorcnt(0);  // this wave's next-chunk DMAs done
    __syncthreads();                       // cross-wave visibility
    p ^= 1;
  }

  // ---- write-out: C/D layout VGPR v -> M = v + 8*(lane>=16), N = lane%16 ----
  const int col   = (int)o0 + wave * 16 + lane16;
  const int mBase = (lane >> 4) * 8;
#pragma unroll
  for (int v = 0; v < 8; ++v) {
    atomicAdd(&y[(long)(mBase + v) * O_DIM + col], acc0[v]);
    atomicAdd(&y[(long)(16 + mBase + v) * O_DIM + col], acc1[v]);
  }
}

extern "C" void kernel_launch(void* const* d_in, const int* in_sizes, int n_in,
                              void* d_out, int out_size, void* d_ws, size_t ws_size,
                              hipStream_t stream) {
  const float* x = (const float*)d_in[0];   // [32, 7168]
  const float* w = (const float*)d_in[1];   // [16384, 7168]
  const float* s = (const float*)d_in[2];   // [128, 56]
  float* y = (float*)d_out;                 // [32, 16384]

  // K-split partials are combined with float atomics -> zero the output first.
  hipMemsetAsync(y, 0, (size_t)out_size * sizeof(float), stream);

  const size_t shmem = (size_t)LDS_FLOATS * sizeof(float); // ~165 KB (WGP has 320 KB)
  hipFuncSetAttribute((const void*)gemm_dq_wmma,
                      hipFuncAttributeMaxDynamicSharedMemorySize, (int)shmem);

  dim3 grid(O_DIM / BLK, KSPLIT);
  gemm_dq_wmma<<<grid, THREADS, shmem, stream>>>(x, w, s, y);
}